// OptionPPO_20968030339598
// MI455X (gfx1250) — compile-verified
//
#include <hip/hip_runtime.h>

// ---------------- problem constants ----------------
#define T_N   65536
#define D_N   512
#define C_N   16
#define A_N   64
#define KST   128        // D_N / 4  (K-steps for 16x16x4 f32 WMMA)
#define GAM   0.99f
#define GT_   (0.99f * 0.95f)
#define ENT_HI_ 0.001f
#define ENT_LO_ 0.001f
#define CHUNK 256
#define NCH   256        // CHUNK*NCH == T_N

typedef float v2f __attribute__((ext_vector_type(2)));
typedef float v8f __attribute__((ext_vector_type(8)));

// LDS layout (floats):
//   [0,8192)        Wc  staged in WMMA-B order   (128 ksteps * 32 lanes * 2)
//   [8192,16384)    Whi staged in WMMA-B order
//   [16384,49152)   Wlo staged in WMMA-B order   (128 ksteps * 4 ntiles * 32 * 2)
//   [49152, +8*1600) per-wave scratch: vc 16x16 (stride 17), hi 16x16 (stride 17),
//                    lo 16x64 (stride 65)  -> bank-conflict-free row reads
#define LDS_BC   0
#define LDS_BH   8192
#define LDS_BL   16384
#define LDS_SCR  49152
#define SCR_PER_WAVE 1600
#define SCR_HI   272            // 16*17
#define SCR_LO   544            // 2*272
#define SMEM_FLOATS (LDS_SCR + 8 * SCR_PER_WAVE)   // 61952
#define SMEM_BYTES  (SMEM_FLOATS * 4)              // 247808 < 320KB

static __device__ __forceinline__ v8f wmma_f32(v2f a, v2f b, v8f c) {
  // V_WMMA_F32_16X16X4_F32: D(16x16,f32) = A(16x4,f32) x B(4x16,f32) + C
  return __builtin_amdgcn_wmma_f32_16x16x4_f32(
      /*neg_a=*/false, a, /*neg_b=*/false, b,
      /*c_mod=*/(short)0, c, /*reuse_a=*/false, /*reuse_b=*/false);
}

__global__ __launch_bounds__(256) void heads_k(
    const float* __restrict__ lo_obs, const float* __restrict__ hi_obs,
    const float* __restrict__ Wc,   const float* __restrict__ Whi,
    const float* __restrict__ Eprev,const float* __restrict__ Wlo,
    const float* __restrict__ Elat, const int* __restrict__ prev_lat,
    const int* __restrict__ latent,
    float* __restrict__ g_vhi, float* __restrict__ g_vlo,
    float* __restrict__ g_enth, float* __restrict__ g_entl)
{
  extern __shared__ float smemF[];
  const int tid = threadIdx.x;

  // ---- stage weights into LDS in WMMA B-operand order ----
  // B tile (4x16) layout: VGPR0: lanes0-15 K=k0 N=lane, lanes16-31 K=k0+2;
  //                       VGPR1: lanes0-15 K=k0+1,       lanes16-31 K=k0+3.
  // LDS element dest index e = ((kk[*NT + nt])*32 + lane)*2 + j, src k = kk*4 + (lane>>4)*2 + j.
  for (int idx = tid; idx < LDS_SCR; idx += 256) {
    if (idx < 8192) {
      int e = idx, j = e & 1, ln = (e >> 1) & 31, kk = e >> 6;
      int k = kk * 4 + (ln >> 4) * 2 + j;
      smemF[LDS_BC + e] = Wc[k * C_N + (ln & 15)];
    } else if (idx < 16384) {
      int e = idx - 8192, j = e & 1, ln = (e >> 1) & 31, kk = e >> 6;
      int k = kk * 4 + (ln >> 4) * 2 + j;
      smemF[LDS_BH + e] = Whi[k * C_N + (ln & 15)];
    } else {
      int e = idx - 16384, j = e & 1, ln = (e >> 1) & 31, nt = (e >> 6) & 3, kk = e >> 8;
      int k = kk * 4 + (ln >> 4) * 2 + j;
      smemF[LDS_BL + e] = Wlo[k * A_N + nt * 16 + (ln & 15)];
    }
  }
  __syncthreads();

  const int wave = tid >> 5, lane = tid & 31, half = lane >> 4, mrow = lane & 15;
  float* scr = smemF + LDS_SCR + wave * SCR_PER_WAVE;
  const int gw = blockIdx.x * 8 + wave;
  const int TW = gridDim.x * 8;

  for (int tile = gw; tile < T_N / 16; tile += TW) {
    const int row = tile * 16 + mrow;
    const float* __restrict__ ph = hi_obs + (size_t)row * D_N + half * 2;
    const float* __restrict__ pl = lo_obs + (size_t)row * D_N + half * 2;

    v8f acc_c = {}, acc_h = {};
    v8f accl[4] = {{}, {}, {}, {}};

    for (int kk = 0; kk < KST; ++kk) {
      // A fragment (16x4 f32): lane m = lane&15, VGPR0/1 = K pair (half*2, half*2+1)
      v2f a  = *(const v2f*)(ph + kk * 4);
      v2f b  = *(const v2f*)(pl + kk * 4);
      v2f bc = *(const v2f*)(smemF + LDS_BC + (kk * 32 + lane) * 2);
      v2f bh = *(const v2f*)(smemF + LDS_BH + (kk * 32 + lane) * 2);
      acc_c = wmma_f32(a, bc, acc_c);
      acc_h = wmma_f32(a, bh, acc_h);
#pragma unroll
      for (int nt = 0; nt < 4; ++nt) {
        v2f bl = *(const v2f*)(smemF + LDS_BL + ((kk * 4 + nt) * 32 + lane) * 2);
        accl[nt] = wmma_f32(b, bl, accl[nt]);
      }
    }

    // ---- spill D tiles to per-wave LDS scratch (row-major, padded) ----
    // D layout: VGPR r, lanes0-15 -> (m=r, n=lane); lanes16-31 -> (m=8+r, n=lane-16)
#pragma unroll
    for (int r = 0; r < 8; ++r) {
      int m = r + half * 8;
      scr[m * 17 + mrow]          = acc_c[r];
      scr[SCR_HI + m * 17 + mrow] = acc_h[r];
#pragma unroll
      for (int nt = 0; nt < 4; ++nt)
        scr[SCR_LO + m * 65 + nt * 16 + mrow] = accl[nt][r];
    }
    __syncthreads();

    // ---- per-row softmax heads: lanes 0..15 each own one row ----
    if (lane < 16) {
      const int t  = tile * 16 + lane;
      const int pv = prev_lat[t], lt = latent[t];
      const float* __restrict__ ep = Eprev + pv * C_N;
      const float* vcr = scr + lane * 17;
      const float* hir = scr + SCR_HI + lane * 17;

      float mx = -3.0e38f;
      for (int n = 0; n < C_N; ++n) mx = fmaxf(mx, hir[n] + ep[n]);
      float s = 0.f, w = 0.f, vh = 0.f;
      for (int n = 0; n < C_N; ++n) {
        float d = hir[n] + ep[n] - mx;
        float e = expf(d);
        s += e; w += d * e; vh += vcr[n] * e;
      }
      float lse = logf(s);
      g_enth[t] = lse - w / s;    // -sum p*logp  (logp = d - lse)
      g_vhi[t]  = vh / s;         // sum vc * p
      g_vlo[t]  = vcr[lt];

      const float* __restrict__ el = Elat + lt * A_N;
      const float* lor = scr + SCR_LO + lane * 65;
      float mx2 = -3.0e38f;
      for (int n = 0; n < A_N; ++n) mx2 = fmaxf(mx2, lor[n] + el[n]);
      float s2 = 0.f, w2 = 0.f;
      for (int n = 0; n < A_N; ++n) {
        float d = lor[n] + el[n] - mx2;
        float e = expf(d);
        s2 += e; w2 += d * e;
      }
      g_entl[t] = logf(s2) - w2 / s2;
    }
    __syncthreads();
  }
}

// ---- GAE pass 1: 256 parallel chunk-local reverse scans (carry-in = 0) ----
__global__ void gae1_k(const float* __restrict__ reward,
                       const float* __restrict__ vhi, const float* __restrict__ vlo,
                       float* __restrict__ ret, float* __restrict__ ah_,
                       float* __restrict__ al_, float* __restrict__ locr,
                       float* __restrict__ locah, float* __restrict__ local_)
{
  const int c = threadIdx.x;
  const int base = c * CHUNK;
  float rs = 0.f, ah = 0.f, al = 0.f;
  for (int t = base + CHUNK - 1; t >= base; --t) {
    float nvh = (t + 1 < T_N) ? vhi[t + 1] : 0.f;   // data lookup, not scan state
    float nvl = (t + 1 < T_N) ? vlo[t + 1] : 0.f;
    float r = reward[t];
    rs = r + GAM * rs;
    ah = (r + GAM * nvh - vhi[t]) + GT_ * ah;
    al = (r + GAM * nvl - vlo[t]) + GT_ * al;
    ret[t] = rs; ah_[t] = ah; al_[t] = al;
  }
  locr[c] = rs; locah[c] = ah; local_[c] = al;
}

// ---- GAE pass 2: single-thread scan over chunk carries ----
__global__ void gae2_k(const float* __restrict__ locr, const float* __restrict__ locah,
                       const float* __restrict__ local_, float* __restrict__ inr,
                       float* __restrict__ inah, float* __restrict__ inal)
{
  const float pr = powf(GAM, (float)CHUNK);
  const float pa = powf(GT_, (float)CHUNK);
  float cr = 0.f, ca = 0.f, cb = 0.f;           // carry into chunk NCH-1 is zero
  for (int c = NCH - 1; c >= 0; --c) {
    inr[c] = cr; inah[c] = ca; inal[c] = cb;
    cr = locr[c]  + pr * cr;
    ca = locah[c] + pa * ca;
    cb = local_[c] + pa * cb;
  }
}

// ---- GAE pass 3: elementwise carry fixup ----
__global__ void gae3_k(float* __restrict__ ret, float* __restrict__ ah_,
                       float* __restrict__ al_, const float* __restrict__ inr,
                       const float* __restrict__ inah, const float* __restrict__ inal)
{
  int t = blockIdx.x * blockDim.x + threadIdx.x;
  if (t >= T_N) return;
  int c = t >> 8;
  float e = (float)((c + 1) * CHUNK - t);        // 1..256
  ret[t] += powf(GAM, e) * inr[c];
  ah_[t] += powf(GT_, e) * inah[c];
  al_[t] += powf(GT_, e) * inal[c];
}

// ---- reduction stage 1: 8 partial sums per block (deterministic) ----
__global__ void red1_k(const float* __restrict__ ah_, const float* __restrict__ al_,
                       const float* __restrict__ vhi, const float* __restrict__ vlo,
                       const float* __restrict__ ret, const float* __restrict__ enth,
                       const float* __restrict__ entl, float* __restrict__ part)
{
  __shared__ float sd[256];
  const int tid = threadIdx.x;
  const int t = blockIdx.x * blockDim.x + tid;
  float a = 0.f, b = 0.f, dh = 0.f, dl = 0.f, eh = 0.f, el = 0.f;
  if (t < T_N) {
    a = ah_[t]; b = al_[t];
    dh = vhi[t] - ret[t]; dl = vlo[t] - ret[t];
    eh = enth[t]; el = entl[t];
  }
  float loc[8] = {a, a * a, b, b * b, dh * dh, dl * dl, eh, el};
  for (int k = 0; k < 8; ++k) {
    sd[tid] = loc[k];
    __syncthreads();
    for (int s = 128; s > 0; s >>= 1) {
      if (tid < s) sd[tid] += sd[tid + s];
      __syncthreads();
    }
    if (tid == 0) part[blockIdx.x * 8 + k] = sd[0];
    __syncthreads();
  }
}

// ---- reduction stage 2 + final loss ----
__global__ void red2_k(const float* __restrict__ part, float* __restrict__ out)
{
  __shared__ float sd[256];
  __shared__ float S[8];
  const int tid = threadIdx.x;
  float loc[8];
  for (int k = 0; k < 8; ++k) loc[k] = part[tid * 8 + k];
  for (int k = 0; k < 8; ++k) {
    sd[tid] = loc[k];
    __syncthreads();
    for (int s = 128; s > 0; s >>= 1) {
      if (tid < s) sd[tid] += sd[tid + s];
      __syncthreads();
    }
    if (tid == 0) S[k] = sd[0];
    __syncthreads();
  }
  if (tid == 0) {
    const float Tf = (float)T_N;
    // ratios r_hi=r_lo=1 exactly (logp - sg(logp) == 0), so
    // pg = -mean(adv_norm) = -((S/T - mu)/(sd+eps)) == -0 exactly.
    float mu_h = S[0] / Tf;
    float var_h = fmaxf((S[1] - S[0] * S[0] / Tf) / (Tf - 1.f), 0.f);
    float pg_h = -((S[0] / Tf) - mu_h) / (sqrtf(var_h) + 1e-8f);
    float mu_l = S[2] / Tf;
    float var_l = fmaxf((S[3] - S[2] * S[2] / Tf) / (Tf - 1.f), 0.f);
    float pg_l = -((S[2] / Tf) - mu_l) / (sqrtf(var_l) + 1e-8f);
    // vpred == vclip == v_f exactly, so vf = mean((v - returns)^2)
    float loss = pg_h + pg_l + 0.5f * ((S[4] + S[5]) / Tf)
               - ENT_HI_ * (S[6] / Tf) - ENT_LO_ * (S[7] / Tf);
    out[0] = loss;
  }
}

extern "C" void kernel_launch(void* const* d_in, const int* in_sizes, int n_in,
                              void* d_out, int out_size, void* d_ws, size_t ws_size,
                              hipStream_t stream) {
  (void)in_sizes; (void)n_in; (void)out_size; (void)ws_size;
  const float* lo_obs = (const float*)d_in[0];
  const float* hi_obs = (const float*)d_in[1];
  const float* reward = (const float*)d_in[2];
  const float* Wc     = (const float*)d_in[3];
  const float* Whi    = (const float*)d_in[4];
  const float* Eprev  = (const float*)d_in[5];
  const float* Wlo    = (const float*)d_in[6];
  const float* Elat   = (const float*)d_in[7];
  // d_in[8] = actions: unused — ratios are exactly 1, logp_lo cancels.
  const int* prev_lat = (const int*)d_in[9];
  const int* latent   = (const int*)d_in[10];

  float* w = (float*)d_ws;
  float* vhi  = w + 0 * T_N;
  float* vlo  = w + 1 * T_N;
  float* enth = w + 2 * T_N;
  float* entl = w + 3 * T_N;
  float* ret  = w + 4 * T_N;
  float* ah   = w + 5 * T_N;
  float* al   = w + 6 * T_N;
  float* locr   = w + 7 * T_N;
  float* locah  = locr + NCH;
  float* local_ = locah + NCH;
  float* inr    = local_ + NCH;
  float* inah   = inr + NCH;
  float* inal   = inah + NCH;
  float* part   = inal + NCH;      // 256 blocks * 8

  (void)hipFuncSetAttribute((const void*)heads_k,
                            hipFuncAttributeMaxDynamicSharedMemorySize, SMEM_BYTES);

  heads_k<<<dim3(128), dim3(256), SMEM_BYTES, stream>>>(
      lo_obs, hi_obs, Wc, Whi, Eprev, Wlo, Elat, prev_lat, latent,
      vhi, vlo, enth, entl);
  gae1_k<<<1, 256, 0, stream>>>(reward, vhi, vlo, ret, ah, al, locr, locah, local_);
  gae2_k<<<1, 1, 0, stream>>>(locr, locah, local_, inr, inah, inal);
  gae3_k<<<dim3(T_N / 256), dim3(256), 0, stream>>>(ret, ah, al, inr, inah, inal);
  red1_k<<<dim3(256), dim3(256), 0, stream>>>(ah, al, vhi, vlo, ret, enth, entl, part);
  red2_k<<<1, 256, 0, stream>>>(part, (float*)d_out);
}